// DWT3D_15908558864455
// MI455X (gfx1250) — compile-verified
//
#include <hip/hip_runtime.h>

// ---------------------------------------------------------------------------
// 3D Haar DWT (single level, separable) for x: (B=2, C=4, D=128, H=256, W=256)
// -> 8 subbands (LLL..HHH), each (2, 4, 64, 128, 128), concatenated in d_out.
//
// Strategy (MI455X / gfx1250, wave32):
//   * Pure streaming problem: 256MB in + 256MB out @ 23.3TB/s => ~22us floor.
//   * One wave handles a 2(D) x 16(H) x 16(W) input block in one pass.
//   * D axis: register butterfly on two loaded planes (P = x0+x1, M = x0-x1).
//   * W axis: Z = T * HaarW^T via V_WMMA_F32_16X16X4_F32, K=16 in 4 chunks.
//       - A operand chunks == the loaded data: per ISA A-layout each lane
//         holds a contiguous float2 of its row -> loads are natural.
//       - B operand == constant Haar matrix chunks (scale 2^-1.5 folded in),
//         built once per wave.
//   * H axis: in the 16x16 f32 C/D layout, H-row pairs (2k,2k+1) are adjacent
//     accumulator VGPRs in identical lanes -> plain v_add/v_sub, no shuffles.
// ---------------------------------------------------------------------------

typedef __attribute__((ext_vector_type(2))) float v2f;
typedef __attribute__((ext_vector_type(8))) float v8f;

// Only meaningful on the device pass; AMDGCN builtins are not declared when
// clang targets the host.
#if defined(__HIP_DEVICE_COMPILE__)
#if !__has_builtin(__builtin_amdgcn_wmma_f32_16x16x4_f32)
#error "missing __builtin_amdgcn_wmma_f32_16x16x4_f32 on this toolchain"
#endif
#endif

// HaarW^T element: Bmat[k][n] = HaarW[n][k], with full 2^-1.5 scale folded in.
//   n<8  (approx row n):  +c at k=2n and k=2n+1
//   n>=8 (detail row n-8): +c at k=2(n-8), -c at k=2(n-8)+1
__device__ __forceinline__ float haarB(int k, int n) {
  const float c = 0.35355339059327379f;  // (1/sqrt(2))^3
  const int nn = n & 7;
  float v = 0.0f;
  if (k == 2 * nn)     v = c;
  if (k == 2 * nn + 1) v = (n < 8) ? c : -c;
  return v;
}

__global__ __launch_bounds__(256) void dwt3d_haar_wmma(
    const float* __restrict__ x, float* __restrict__ out) {
  const int lane = threadIdx.x & 31;
  const int wave = threadIdx.x >> 5;
  const int blk  = blockIdx.x * 8 + wave;   // [0, 131072)

  const int half = lane >> 4;  // 0: lanes 0-15, 1: lanes 16-31
  const int n16  = lane & 15;

  // --- constant B operand (HaarW^T) in assumed 4x16 K-major B layout:
  // VGPR0 holds K=4j+2*half, VGPR1 holds K=4j+2*half+1, N = lane&15.
  v2f b[4];
#pragma unroll
  for (int j = 0; j < 4; ++j) {
    const int k = 4 * j + 2 * half;
    b[j].x = haarB(k, n16);
    b[j].y = haarB(k + 1, n16);
  }

  // --- block coordinates: blk = ((bc*64 + d2)*16 + ht)*16 + wt
  const int wt = blk & 15;
  const int ht = (blk >> 4) & 15;
  const int d2 = (blk >> 8) & 63;
  const int bc = blk >> 14;                 // b*C + c in [0,8)

  // --- input addressing: x is (8, 128, 256, 256) flat.
  // A-layout: lane holds row m = n16, cols (4j + 2*half) and (4j + 2*half + 1).
  const int planeBase = ((bc * 128 + 2 * d2) * 256 + ht * 16) * 256 + wt * 16;
  const float* p0 = x + planeBase + n16 * 256 + 2 * half;
  const float* p1 = p0 + 65536;             // next D plane (H*W)

  // --- load both D planes in A-chunk layout; D-axis butterfly in registers.
  v2f P[4], M[4];
#pragma unroll
  for (int j = 0; j < 4; ++j) {
    const v2f a0 = *(const v2f*)(p0 + 4 * j);   // global_load_b64
    const v2f a1 = *(const v2f*)(p1 + 4 * j);
    P[j] = a0 + a1;                             // -> D approx band
    M[j] = a0 - a1;                             // -> D detail band
  }

  // --- W-axis transform: Y = T * HaarW^T, K=16 accumulated over 4 chunks.
  v8f yL = {};
  v8f yH = {};
#pragma unroll
  for (int j = 0; j < 4; ++j) {
    yL = __builtin_amdgcn_wmma_f32_16x16x4_f32(
        false, P[j], false, b[j], (short)0, yL, false, false);
    yH = __builtin_amdgcn_wmma_f32_16x16x4_f32(
        false, M[j], false, b[j], (short)0, yH, false, false);
  }

  // --- output addressing.
  // Subband s = 4*dBand + 2*hBand + wBand, each subband 8*64*128*128 elems.
  const int SB = 8 * 64 * 128 * 128;        // 8388608
  const int rowBase = (bc * 64 + d2) * 128 + ht * 8;
  const int wB = n16 >> 3;                  // W band from output column half
  // lane-dependent part: W-band stride + H-row half offset + column
  const int laneOut = wB * SB + (half * 4) * 128 + (lane & 7);
  float* obase = out + rowBase * 128 + wt * 8 + laneOut;

  // --- H-axis butterfly directly on accumulator VGPRs + stores.
  // acc[r]: lanes0-15 = Z[row r], lanes16-31 = Z[row 8+r]; pairs (2k,2k+1)
  // are adjacent VGPRs in the same lanes. Output H row = k + 4*half.
#pragma unroll
  for (int k = 0; k < 4; ++k) {
    const float sL = yL[2 * k] + yL[2 * k + 1];  // D=L, H=L
    const float dL = yL[2 * k] - yL[2 * k + 1];  // D=L, H=H
    const float sH = yH[2 * k] + yH[2 * k + 1];  // D=H, H=L
    const float dH = yH[2 * k] - yH[2 * k + 1];  // D=H, H=H
    float* o = obase + k * 128;
    o[0]          = sL;   // subband (0,0,wB)
    o[2 * SB]     = dL;   // subband (0,1,wB)
    o[4 * SB]     = sH;   // subband (1,0,wB)
    o[6 * SB]     = dH;   // subband (1,1,wB)
  }
}

extern "C" void kernel_launch(void* const* d_in, const int* in_sizes, int n_in,
                              void* d_out, int out_size, void* d_ws, size_t ws_size,
                              hipStream_t stream) {
  (void)in_sizes; (void)n_in; (void)out_size; (void)d_ws; (void)ws_size;
  const float* x = (const float*)d_in[0];
  float* out = (float*)d_out;
  // 131072 wave-blocks / 8 waves per 256-thread workgroup = 16384 workgroups.
  dwt3d_haar_wmma<<<dim3(16384), dim3(256), 0, stream>>>(x, out);
}